// GraphTransformer_74448963108869
// MI455X (gfx1250) — compile-verified
//
#include <hip/hip_runtime.h>
#include <math.h>

typedef float v2f __attribute__((ext_vector_type(2)));
typedef float v8f __attribute__((ext_vector_type(8)));

// ---------------------------------------------------------------------------
// Utility: zero a float region
// ---------------------------------------------------------------------------
__global__ void k_zero(float* __restrict__ p, int n) {
    int i = blockIdx.x * blockDim.x + threadIdx.x;
    if (i < n) p[i] = 0.0f;
}

// ---------------------------------------------------------------------------
// Monotone float <-> uint key for atomicMax-based segment max
// ---------------------------------------------------------------------------
__device__ __forceinline__ unsigned fkey(float f) {
    unsigned u = __float_as_uint(f);
    return (u & 0x80000000u) ? ~u : (u | 0x80000000u);
}
__device__ __forceinline__ float funkey(unsigned k) {
    unsigned u = (k & 0x80000000u) ? (k & 0x7FFFFFFFu) : ~k;
    return __uint_as_float(u);
}

// ---------------------------------------------------------------------------
// FP32 WMMA GEMM (full tiles): C[M,Nc] = A[M,K] @ W[K,Nc] + bias, row-major.
// Requires M%16==0, K%4==0, Nc%32==0. One wave computes a 16x32 tile
// (two V_WMMA_F32_16X16X4_F32 accumulators sharing one A fragment).
// A frag (16x4): lanes 0-15 -> M=lane, VGPRs = K=k0,k0+1 ; lanes 16-31 -> K=k0+2,k0+3
// B frag (4x16): lanes 0-15 -> N=lane, VGPRs = K=k0,k0+1 ; lanes 16-31 -> K=k0+2,k0+3
// D (16x16):     VGPR r: lanes 0-15 -> row r, col=lane ; lanes 16-31 -> row r+8
// ---------------------------------------------------------------------------
__global__ void k_gemm_wmma_full(const float* __restrict__ A, int lda,
                                 const float* __restrict__ W, int ldw,
                                 const float* __restrict__ bias,
                                 float* __restrict__ C, int ldc,
                                 int K, int Nc) {
    int tileM = blockIdx.x;
    int pair  = blockIdx.y * 4 + threadIdx.y;    // each pair = 32 output columns
    if (pair >= (Nc >> 5)) return;               // wave-uniform exit

    int lane = threadIdx.x;
    int half = lane >> 4;
    int l16  = lane & 15;
    int rowA = tileM * 16 + l16;
    int col0 = pair * 32 + l16;
    int col1 = col0 + 16;

    const float* Arow = A + (size_t)rowA * lda + 2 * half;
    const float* Wh   = W + (size_t)(2 * half) * ldw;
    v8f acc0 = {}, acc1 = {};
    for (int k0 = 0; k0 < K; k0 += 4) {
        v2f a, b0, b1;
        a.x = Arow[k0];
        a.y = Arow[k0 + 1];
        const float* w0 = Wh + (size_t)k0 * ldw;
        b0.x = w0[col0];
        b0.y = w0[ldw + col0];
        b1.x = w0[col1];
        b1.y = w0[ldw + col1];
        acc0 = __builtin_amdgcn_wmma_f32_16x16x4_f32(
            false, a, false, b0, (short)0, acc0, false, false);
        acc1 = __builtin_amdgcn_wmma_f32_16x16x4_f32(
            false, a, false, b1, (short)0, acc1, false, false);
    }
    float bv0 = bias ? bias[col0] : 0.0f;
    float bv1 = bias ? bias[col1] : 0.0f;
    int rbase = tileM * 16 + half * 8;
    #pragma unroll
    for (int r = 0; r < 8; ++r) {
        float* crow = C + (size_t)(rbase + r) * ldc;
        crow[col0] = acc0[r] + bv0;
        crow[col1] = acc1[r] + bv1;
    }
}

// ---------------------------------------------------------------------------
// FP32 WMMA GEMM (column-masked): used only for tiny Nc (layer 2, Nc=6).
// ---------------------------------------------------------------------------
__global__ void k_gemm_wmma_masked(const float* __restrict__ A, int lda,
                                   const float* __restrict__ W, int ldw,
                                   const float* __restrict__ bias,
                                   float* __restrict__ C, int ldc,
                                   int K, int Nc) {
    int tileM = blockIdx.x;
    int tileN = blockIdx.y * 4 + threadIdx.y;
    if (tileN >= ((Nc + 15) >> 4)) return;       // wave-uniform exit

    int lane = threadIdx.x;
    int half = lane >> 4;
    int l16  = lane & 15;
    int rowA = tileM * 16 + l16;
    int colB = tileN * 16 + l16;
    bool colOK = colB < Nc;

    const float* Arow = A + (size_t)rowA * lda + 2 * half;
    v8f acc = {};
    for (int k0 = 0; k0 < K; k0 += 4) {
        v2f a, b;
        a.x = Arow[k0];
        a.y = Arow[k0 + 1];
        int kb = k0 + 2 * half;
        float b0 = 0.0f, b1 = 0.0f;
        if (colOK) {
            b0 = W[(size_t)kb * ldw + colB];
            b1 = W[(size_t)(kb + 1) * ldw + colB];
        }
        b.x = b0; b.y = b1;
        acc = __builtin_amdgcn_wmma_f32_16x16x4_f32(
            false, a, false, b, (short)0, acc, false, false);
    }
    if (colOK) {
        float bv = bias ? bias[colB] : 0.0f;
        int rbase = tileM * 16 + half * 8;
        #pragma unroll
        for (int r = 0; r < 8; ++r)
            C[(size_t)(rbase + r) * ldc + colB] = acc[r] + bv;
    }
}

// ---------------------------------------------------------------------------
// G[n][h][d] = sum_c Q[n][h*C+c] * we[d][h*C+c]   (q projected onto edge basis)
// ---------------------------------------------------------------------------
__global__ void k_compute_g(const float* __restrict__ Q, const float* __restrict__ we,
                            float* __restrict__ G, int n, int H, int Cc, int HC) {
    int tid = blockIdx.x * blockDim.x + threadIdx.x;
    if (tid >= n * H * 16) return;
    int d = tid & 15;
    int t = tid >> 4;
    int h = t % H;
    int nd = t / H;
    const float* q = Q + (size_t)nd * HC + h * Cc;
    const float* w = we + (size_t)d * HC + h * Cc;
    float s = 0.0f;
    for (int c = 0; c < Cc; ++c) s += q[c] * w[c];
    G[(size_t)nd * H * 16 + h * 16 + d] = s;
}

// ---------------------------------------------------------------------------
// Pass B: alpha[e,h] = (q[dst]·k[src] + G[dst]·eattr[e]) / sqrt(C); segment max
// ---------------------------------------------------------------------------
__global__ void k_edge_alpha(const float* __restrict__ Q, const float* __restrict__ Kf,
                             const float* __restrict__ G, const float* __restrict__ ea,
                             const int* __restrict__ ei, float* __restrict__ ALPHA,
                             unsigned* __restrict__ Mk,
                             int E, int H, int Cc, int HC, float rs) {
    int tid = blockIdx.x * blockDim.x + threadIdx.x;
    if (tid >= E * H) return;
    int h = tid % H, e = tid / H;
    int src = ei[e], dst = ei[E + e];
    const float* q = Q  + (size_t)dst * HC + h * Cc;
    const float* k = Kf + (size_t)src * HC + h * Cc;
    float s = 0.0f;
    for (int c = 0; c < Cc; ++c) s += q[c] * k[c];
    const float* g = G + (size_t)dst * H * 16 + h * 16;
    const float* a = ea + (size_t)e * 16;
    #pragma unroll
    for (int d = 0; d < 16; ++d) s += g[d] * a[d];
    float alpha = s * rs;
    ALPHA[tid] = alpha;
    atomicMax(&Mk[dst * H + h], fkey(alpha));
}

// ---------------------------------------------------------------------------
// Pass C: ex = exp(alpha - m); den[dst,h] += ex
// ---------------------------------------------------------------------------
__global__ void k_edge_exp(float* __restrict__ ALPHA, const unsigned* __restrict__ Mk,
                           float* __restrict__ DEN, const int* __restrict__ ei,
                           int E, int H) {
    int tid = blockIdx.x * blockDim.x + threadIdx.x;
    if (tid >= E * H) return;
    int h = tid % H, e = tid / H;
    int dst = ei[E + e];
    float m = funkey(Mk[dst * H + h]);
    if (!isfinite(m)) m = 0.0f;
    float ex = expf(ALPHA[tid] - m);
    ALPHA[tid] = ex;
    atomicAdd(&DEN[dst * H + h], ex);
}

// ---------------------------------------------------------------------------
// Pass D: out[dst] += att * v[src];  SB[dst,h,:] += att * eattr[e,:]
// ---------------------------------------------------------------------------
__global__ void k_edge_agg(const float* __restrict__ ALPHA, const float* __restrict__ DEN,
                           const float* __restrict__ V, const float* __restrict__ ea,
                           const int* __restrict__ ei,
                           float* __restrict__ OUT, float* __restrict__ SB,
                           int E, int H, int Cc, int HC) {
    int tid = blockIdx.x * blockDim.x + threadIdx.x;
    if (tid >= E * H) return;
    int h = tid % H, e = tid / H;
    int src = ei[e], dst = ei[E + e];
    float att = ALPHA[tid] / DEN[dst * H + h];
    float* o = OUT + (size_t)dst * HC + h * Cc;
    const float* v = V + (size_t)src * HC + h * Cc;
    for (int c = 0; c < Cc; ++c) atomicAdd(&o[c], att * v[c]);
    float* sb = SB + (size_t)dst * H * 16 + h * 16;
    const float* a = ea + (size_t)e * 16;
    #pragma unroll
    for (int d = 0; d < 16; ++d) atomicAdd(&sb[d], att * a[d]);
}

// ---------------------------------------------------------------------------
// Pass E: out_j += SB·we ; beta gate (wave32 reduction) ; BN statistics
// One wave per node. dout == H*C for all layers (H==1 => mean == identity).
// ---------------------------------------------------------------------------
__global__ void k_finalize(const float* __restrict__ OUT, const float* __restrict__ SB,
                           const float* __restrict__ S, const float* __restrict__ we,
                           const float* __restrict__ wbeta,
                           float* __restrict__ Xn, float* __restrict__ BN,
                           int n, int H, int Cc, int HC, int hasBN) {
    int node = blockIdx.x * 4 + threadIdx.y;
    if (node >= n) return;
    int lane = threadIdx.x;
    int dout = HC;
    int nt = (dout + 31) >> 5;
    float outv[4], xrv[4];
    float part = 0.0f;
    for (int t = 0; t < nt; ++t) {
        int j = t * 32 + lane;
        float o = 0.0f, xr = 0.0f;
        if (j < dout) {
            int h = j / Cc;
            o = OUT[(size_t)node * HC + j];
            const float* sb = SB + (size_t)node * H * 16 + h * 16;
            const float* w  = we + j;
            #pragma unroll
            for (int d = 0; d < 16; ++d) o += sb[d] * w[(size_t)d * HC];
            xr = S[(size_t)node * dout + j];
            part += wbeta[j] * o + wbeta[dout + j] * xr + wbeta[2 * dout + j] * (o - xr);
        }
        outv[t] = o; xrv[t] = xr;
    }
    for (int off = 16; off > 0; off >>= 1) part += __shfl_xor(part, off, 32);
    float b = 1.0f / (1.0f + expf(-part));
    for (int t = 0; t < nt; ++t) {
        int j = t * 32 + lane;
        if (j < dout) {
            float xv = b * xrv[t] + (1.0f - b) * outv[t];
            Xn[(size_t)node * dout + j] = xv;
            if (hasBN) {
                atomicAdd(&BN[j], xv);
                atomicAdd(&BN[128 + j], xv * xv);
            }
        }
    }
}

// ---------------------------------------------------------------------------
// Pass F: BatchNorm (batch stats) + ReLU, in place
// ---------------------------------------------------------------------------
__global__ void k_bn_relu(float* __restrict__ X, const float* __restrict__ BN,
                          const float* __restrict__ gamma, const float* __restrict__ beta,
                          int n, int dout) {
    int tid = blockIdx.x * blockDim.x + threadIdx.x;
    if (tid >= n * dout) return;
    int j = tid % dout;
    float inv_n = 1.0f / (float)n;
    float mean = BN[j] * inv_n;
    float var  = BN[128 + j] * inv_n - mean * mean;
    float x = (X[tid] - mean) * rsqrtf(var + 1e-5f) * gamma[j] + beta[j];
    X[tid] = fmaxf(x, 0.0f);
}

// ---------------------------------------------------------------------------
// Host orchestration
// ---------------------------------------------------------------------------
extern "C" void kernel_launch(void* const* d_in, const int* in_sizes, int n_in,
                              void* d_out, int out_size, void* d_ws, size_t ws_size,
                              hipStream_t stream) {
    const float* x0  = (const float*)d_in[0];
    const int*   ei  = (const int*)d_in[1];
    const float* ea  = (const float*)d_in[2];
    const int Nn = in_sizes[0] / 64;    // 50000 (divisible by 16)
    const int E  = in_sizes[2] / 16;    // 400000

    float* ws = (float*)d_ws;
    size_t off = 0;
    auto alloc = [&](size_t cnt) { float* p = ws + off; off += cnt; return p; };
    float* X0    = alloc((size_t)Nn * 128);
    float* X1    = alloc((size_t)Nn * 128);
    float* Q     = alloc((size_t)Nn * 128);
    float* Kb    = alloc((size_t)Nn * 128);
    float* V     = alloc((size_t)Nn * 128);
    float* S     = alloc((size_t)Nn * 128);
    float* G     = alloc((size_t)Nn * 64);
    float* ALPHA = alloc((size_t)E * 4);
    // contiguous zero region: SB | Mk | DEN | BN
    float* SB    = alloc((size_t)Nn * 64);
    float* Mk    = alloc((size_t)Nn * 4);
    float* DEN   = alloc((size_t)Nn * 4);
    float* BN    = alloc(256);
    const int zeroCnt = Nn * 64 + Nn * 4 + Nn * 4 + 256;
    float* OUT = Q;   // Q is dead after pass B -> reuse as aggregation buffer

    struct Cfg { int din, H, C, hasBN, pbase; };
    const Cfg cfgs[3] = { {64, 4, 32, 1, 3}, {128, 4, 32, 1, 15}, {128, 1, 6, 0, 27} };

    const float* xin = x0;
    for (int i = 0; i < 3; ++i) {
        const Cfg& cf = cfgs[i];
        const int H = cf.H, Cc = cf.C, din = cf.din;
        const int HC = H * Cc, dout = HC;
        void* const* p = d_in + cf.pbase;
        const float* wq = (const float*)p[0]; const float* bq = (const float*)p[1];
        const float* wk = (const float*)p[2]; const float* bk = (const float*)p[3];
        const float* wv = (const float*)p[4]; const float* bv = (const float*)p[5];
        const float* we = (const float*)p[6];
        const float* wsk = (const float*)p[7]; const float* bsk = (const float*)p[8];
        const float* wbeta = (const float*)p[9];
        float* xout = (i == 0) ? X0 : (i == 1) ? X1 : (float*)d_out;

        // --- node GEMMs (fp32 WMMA) ---
        if ((HC & 31) == 0) {
            int pairs = HC >> 5;
            dim3 blk(32, 4), grd(Nn / 16, (pairs + 3) / 4);
            k_gemm_wmma_full<<<grd, blk, 0, stream>>>(xin, din, wq, HC, bq, Q,  HC, din, HC);
            k_gemm_wmma_full<<<grd, blk, 0, stream>>>(xin, din, wk, HC, bk, Kb, HC, din, HC);
            k_gemm_wmma_full<<<grd, blk, 0, stream>>>(xin, din, wv, HC, bv, V,  HC, din, HC);
            k_gemm_wmma_full<<<grd, blk, 0, stream>>>(xin, din, wsk, dout, bsk, S, dout, din, dout);
        } else {
            int tilesN = (HC + 15) >> 4;
            dim3 blk(32, 4), grd(Nn / 16, (tilesN + 3) / 4);
            k_gemm_wmma_masked<<<grd, blk, 0, stream>>>(xin, din, wq, HC, bq, Q,  HC, din, HC);
            k_gemm_wmma_masked<<<grd, blk, 0, stream>>>(xin, din, wk, HC, bk, Kb, HC, din, HC);
            k_gemm_wmma_masked<<<grd, blk, 0, stream>>>(xin, din, wv, HC, bv, V,  HC, din, HC);
            k_gemm_wmma_masked<<<grd, blk, 0, stream>>>(xin, din, wsk, dout, bsk, S, dout, din, dout);
        }
        // --- G = Q projected onto edge basis ---
        {
            int tot = Nn * H * 16;
            k_compute_g<<<(tot + 255) / 256, 256, 0, stream>>>(Q, we, G, Nn, H, Cc, HC);
        }
        // --- zero accumulators (SB, Mk=-inf key 0, DEN, BN stats) ---
        k_zero<<<(zeroCnt + 255) / 256, 256, 0, stream>>>(SB, zeroCnt);
        // --- edge attention ---
        {
            int tot = E * H;
            float rs = 1.0f / sqrtf((float)Cc);
            k_edge_alpha<<<(tot + 255) / 256, 256, 0, stream>>>(
                Q, Kb, G, ea, ei, ALPHA, (unsigned*)Mk, E, H, Cc, HC, rs);
            k_zero<<<(Nn * HC + 255) / 256, 256, 0, stream>>>(OUT, Nn * HC);
            k_edge_exp<<<(tot + 255) / 256, 256, 0, stream>>>(
                ALPHA, (const unsigned*)Mk, DEN, ei, E, H);
            k_edge_agg<<<(tot + 255) / 256, 256, 0, stream>>>(
                ALPHA, DEN, V, ea, ei, OUT, SB, E, H, Cc, HC);
        }
        // --- finalize: edge-basis expansion, beta gate, BN stats ---
        k_finalize<<<(Nn + 3) / 4, dim3(32, 4), 0, stream>>>(
            OUT, SB, S, we, wbeta, xout, BN, Nn, H, Cc, HC, cf.hasBN);
        if (cf.hasBN) {
            const float* gamma = (const float*)p[10];
            const float* beta  = (const float*)p[11];
            k_bn_relu<<<(Nn * dout + 255) / 256, 256, 0, stream>>>(
                xout, BN, gamma, beta, Nn, dout);
        }
        xin = xout;
    }
    (void)n_in; (void)out_size; (void)ws_size;
}